// GraphConvolution_66984309948593
// MI455X (gfx1250) — compile-verified
//
#include <hip/hip_runtime.h>
#include <hip/hip_bf16.h>

typedef __attribute__((ext_vector_type(2))) float v2f;
typedef __attribute__((ext_vector_type(4))) float v4f;
typedef __attribute__((ext_vector_type(8))) float v8f;

#define FDIM   128
#define KDIM   3
#define SUPW   384   // KDIM * FDIM

#define WMMA_F32(A, B, C) \
    __builtin_amdgcn_wmma_f32_16x16x4_f32(false, (A), false, (B), (short)0, (C), false, false)

// ---------------------------------------------------------------------------
// Kernel 1: repack weight (F_IN, F_OUT, K) into WMMA B-fragment order.
// Logical B matrix: B[i, j] with i = F_IN row, j = k*128 + f  (128 x 384).
// Fragment layout: for column group g (16 cols) and k-step kk (K base 4*kk),
//   Wp[(g*32 + kk)*64 + lane*2 + jj] = B[4*kk + 2*(lane>>4) + jj, 16*g + (lane&15)]
// so each lane's (b.x, b.y) is one contiguous b64, coalesced across the wave.
// ---------------------------------------------------------------------------
__global__ void gc_repack_weight(const float* __restrict__ w,
                                 float* __restrict__ wp) {
    int p = blockIdx.x * blockDim.x + threadIdx.x;      // 0 .. 128*384-1
    int jj   = p & 1;
    int lane = (p >> 1) & 31;
    int kk   = (p >> 6) & 31;
    int g    = p >> 11;                                  // 0..23
    int krow = 4 * kk + 2 * (lane >> 4) + jj;            // F_IN index 0..127
    int jcol = 16 * g + (lane & 15);                     // 0..383
    int f = jcol & 127;                                  // F_OUT index
    int k = jcol >> 7;                                   // kernel index
    wp[p] = w[krow * SUPW + f * KDIM + k];
}

// ---------------------------------------------------------------------------
// Kernel 2: support = X (N x 128) @ B (128 x 384), fp32 WMMA 16x16x4.
// One wave -> 32 rows x 64 cols (2 A fragments x 4 B tiles, 8 accumulators).
// Per kk-step: 2 A b64 loads + 4 B b64 loads -> 8 v_wmma.
// ---------------------------------------------------------------------------
__global__ __launch_bounds__(256) void gc_gemm_wmma(const float* __restrict__ X,
                                                    const float* __restrict__ Wp,
                                                    float* __restrict__ S,
                                                    int rowTiles, int N_) {
    const int lane = threadIdx.x & 31;
    const int wave = threadIdx.x >> 5;
    const int rt0  = (blockIdx.x * 8 + wave) * 2;
    if (rt0 >= rowTiles) return;                         // wave-uniform exit
    const bool rt1v = (rt0 + 1) < rowTiles;

    const int m0 = rt0 << 4;                             // rows m0..m0+15
    const int m1 = rt1v ? (m0 + 16) : (N_ - 16);         // clamp tail (loads only)
    const int g0 = blockIdx.y * 4;                       // 4 x 16-col tiles
    const int r    = lane & 15;
    const int half = lane >> 4;

    // A fragment (16x4): lane(r,half) holds X[m + r, 4*kk + 2*half + {0,1}]
    const float* xrow0 = X + (size_t)(m0 + r) * FDIM + 2 * half;
    const float* xrow1 = X + (size_t)(m1 + r) * FDIM + 2 * half;
    // B fragments pre-swizzled: one b64 per lane per (tile, kk)
    const float* bbase = Wp + (size_t)g0 * 32 * 64 + lane * 2;

    v8f c00 = {}, c01 = {}, c02 = {}, c03 = {};
    v8f c10 = {}, c11 = {}, c12 = {}, c13 = {};

#pragma unroll 4
    for (int kk = 0; kk < 32; ++kk) {                    // K = 128 = 32 * 4
        v2f a0 = *(const v2f*)(xrow0 + 4 * kk);
        v2f a1 = *(const v2f*)(xrow1 + 4 * kk);
        v2f b0 = *(const v2f*)(bbase + (size_t)(0 * 32 + kk) * 64);
        v2f b1 = *(const v2f*)(bbase + (size_t)(1 * 32 + kk) * 64);
        v2f b2 = *(const v2f*)(bbase + (size_t)(2 * 32 + kk) * 64);
        v2f b3 = *(const v2f*)(bbase + (size_t)(3 * 32 + kk) * 64);
        c00 = WMMA_F32(a0, b0, c00);
        c01 = WMMA_F32(a0, b1, c01);
        c02 = WMMA_F32(a0, b2, c02);
        c03 = WMMA_F32(a0, b3, c03);
        c10 = WMMA_F32(a1, b0, c10);
        c11 = WMMA_F32(a1, b1, c11);
        c12 = WMMA_F32(a1, b2, c12);
        c13 = WMMA_F32(a1, b3, c13);
    }

    // C/D layout: VGPR i -> row m + half*8 + i, col colBase + 16*t + r
    const int colBase = g0 << 4;
    float* d0 = S + (size_t)(m0 + half * 8) * SUPW + colBase + r;
#pragma unroll
    for (int i = 0; i < 8; ++i) {
        d0[(size_t)i * SUPW +  0] = c00[i];
        d0[(size_t)i * SUPW + 16] = c01[i];
        d0[(size_t)i * SUPW + 32] = c02[i];
        d0[(size_t)i * SUPW + 48] = c03[i];
    }
    if (rt1v) {                                          // avoid racing tail rows
        float* d1 = S + (size_t)(m1 + half * 8) * SUPW + colBase + r;
#pragma unroll
        for (int i = 0; i < 8; ++i) {
            d1[(size_t)i * SUPW +  0] = c10[i];
            d1[(size_t)i * SUPW + 16] = c11[i];
            d1[(size_t)i * SUPW + 32] = c12[i];
            d1[(size_t)i * SUPW + 48] = c13[i];
        }
    }
}

// ---------------------------------------------------------------------------
// Kernel 3: out[n, f] = bias[f]
// ---------------------------------------------------------------------------
__global__ void gc_bias_init(const float* __restrict__ bias,
                             float* __restrict__ out, long long total) {
    long long i = (long long)blockIdx.x * blockDim.x + threadIdx.x;
    if (i < total) out[i] = bias[i & (FDIM - 1)];
}

// ---------------------------------------------------------------------------
// Kernel 4: one wave per edge.
// acc[f] = sum_k TT[e,k] * support[col, k*128 + f];  atomic out[row, f] += acc
// Lane l handles floats 4l..4l+3 -> contiguous 512B gathers per k-slice.
// ---------------------------------------------------------------------------
__global__ __launch_bounds__(256) void gc_edge_scatter(const long long* __restrict__ eidx,
                                                       const float* __restrict__ TT,
                                                       const float* __restrict__ support,
                                                       float* __restrict__ out, int E_) {
    int wave = (int)((blockIdx.x * (long long)blockDim.x + threadIdx.x) >> 5);
    int lane = threadIdx.x & 31;
    if (wave >= E_) return;

    int row = (int)eidx[wave];
    int col = (int)eidx[(size_t)E_ + wave];
    float t0 = TT[(size_t)wave * KDIM + 0];
    float t1 = TT[(size_t)wave * KDIM + 1];
    float t2 = TT[(size_t)wave * KDIM + 2];

    const v4f* sp = (const v4f*)(support + (size_t)col * SUPW);
    v4f acc = t0 * sp[lane] + t1 * sp[32 + lane] + t2 * sp[64 + lane];

    float* o = out + (size_t)row * FDIM + 4 * lane;
    unsafeAtomicAdd(o + 0, acc.x);
    unsafeAtomicAdd(o + 1, acc.y);
    unsafeAtomicAdd(o + 2, acc.z);
    unsafeAtomicAdd(o + 3, acc.w);
}

// ---------------------------------------------------------------------------
extern "C" void kernel_launch(void* const* d_in, const int* in_sizes, int n_in,
                              void* d_out, int out_size, void* d_ws, size_t ws_size,
                              hipStream_t stream) {
    const float*     x      = (const float*)d_in[0];
    const long long* eidx   = (const long long*)d_in[1];   // int64 (2, E)
    const float*     TT     = (const float*)d_in[2];
    const float*     weight = (const float*)d_in[3];
    const float*     bias   = (const float*)d_in[4];
    float*           out    = (float*)d_out;

    const int N_ = in_sizes[0] / FDIM;     // 50000
    const int E_ = in_sizes[1] / 2;        // 800000

    // workspace: [support N x 384 | Wp 128 x 384 fragment-packed]
    float* support = (float*)d_ws;
    float* wp      = support + (size_t)N_ * SUPW;

    // 1) weight repack into WMMA fragment order (128*384 elements)
    gc_repack_weight<<<(FDIM * SUPW) / 256, 256, 0, stream>>>(weight, wp);

    // 2) WMMA GEMM: 32-row wave tiles x 6 column groups of 64
    int rowTiles  = (N_ + 15) / 16;                      // 3125
    int waveTiles = (rowTiles + 1) / 2;                  // 1563
    dim3 g((waveTiles + 7) / 8, SUPW / 64);
    gc_gemm_wmma<<<g, 256, 0, stream>>>(x, wp, support, rowTiles, N_);

    // 3) bias broadcast into out
    long long total = (long long)N_ * FDIM;
    gc_bias_init<<<(int)((total + 255) / 256), 256, 0, stream>>>(bias, out, total);

    // 4) edge gather + fused-K scale + atomic scatter (1 wave / edge)
    gc_edge_scatter<<<(E_ + 7) / 8, 256, 0, stream>>>(eidx, TT, support, out, E_);
}